// GCNLayerOrig_670014899156
// MI455X (gfx1250) — compile-verified
//
#include <hip/hip_runtime.h>

// GCN layer for MI455X (gfx1250), fp32 WMMA path with async direct-to-LDS
// double-buffered pipelines.
//   dinv  = rsqrt(rowsum(adj_hat) - diag(adj_hat))          [B,N]
//   hs    = dinv[m] * (x @ W^T + b)                          [B,N,D]
//   out   = dinv[n] * (adj_hat @ hs)                         [B,N,D]
// Workspace layout (floats): dinv[B*N] | Wt[D*D] | hs[B*N*D]  (~17.1 MB total)

typedef __attribute__((ext_vector_type(2))) float v2f;
typedef __attribute__((ext_vector_type(8))) float v8f;

#define BATCH 8
#define NN    2048
#define DD    256

#define KT   16    // K-slice per pipeline stage
#define LDA  18    // LDS row stride (floats) for 128xKT A tiles
#define LDB  136   // LDS row stride (floats) for KTx128 B tiles
#define ASZ  (128 * LDA)
#define BSZ  (KT * LDB)

static __device__ __forceinline__ v8f wmma_f32_16x16x4(v2f a, v2f b, v8f c) {
  // (neg_a, A, neg_b, B, c_mod, C, reuse_a, reuse_b)
  return __builtin_amdgcn_wmma_f32_16x16x4_f32(false, a, false, b, (short)0, c,
                                               false, false);
}

// Async global->LDS copy, 8 bytes per lane, tracked by ASYNCcnt.
// LDS byte offset = low 32 bits of the generic pointer.
static __device__ __forceinline__ void async_copy_b64(float* lds_dst,
                                                      const float* gsrc) {
  const unsigned lds_off = (unsigned)(uintptr_t)lds_dst;
  asm volatile("global_load_async_to_lds_b64 %0, %1, off"
               :: "v"(lds_off), "v"((unsigned long long)(uintptr_t)gsrc)
               : "memory");
}

// ---------------------------------------------------------------- degrees ---
// One wave per adjacency row (B*N rows of length N).
__global__ __launch_bounds__(256) void gcn_degree_kernel(
    const float* __restrict__ adj, float* __restrict__ dinv) {
  const int lane = threadIdx.x;                            // 0..31
  const int row  = blockIdx.x * blockDim.y + threadIdx.y;  // 0..B*N-1
  const float* __restrict__ rp = adj + (size_t)row * NN;
  const float4* __restrict__ rp4 = (const float4*)rp;
  float s = 0.0f;
  for (int i = lane; i < NN / 4; i += 32) {
    float4 v = rp4[i];
    s += v.x + v.y + v.z + v.w;
  }
  for (int off = 16; off; off >>= 1) s += __shfl_xor(s, off, 32);
  if (lane == 0) {
    float d = s - rp[row & (NN - 1)];                      // subtract diagonal
    dinv[row] = rsqrtf(d);
  }
}

// ------------------------------------------------------------ W transpose ---
__global__ __launch_bounds__(256) void gcn_transpose_kernel(
    const float* __restrict__ W, float* __restrict__ Wt) {
  __shared__ float t[32][33];
  const int bx = blockIdx.x * 32, by = blockIdx.y * 32;
  const int x = threadIdx.x, y = threadIdx.y;              // 32x8
#pragma unroll
  for (int i = 0; i < 32; i += 8)
    t[y + i][x] = W[(size_t)(by + y + i) * DD + bx + x];
  __syncthreads();
#pragma unroll
  for (int i = 0; i < 32; i += 8)
    Wt[(size_t)(bx + y + i) * DD + by + x] = t[x][y + i];
}

// ------------------------------------------------- shared GEMM micro-core ---
// 8 waves; wave (wy,wx) computes rows mw..mw+31, cols nw..nw+63 of a 128x128
// tile via v_wmma_f32_16x16x4_f32.  A frag: lanes 0-15 K={0,1}, 16-31 K={2,3}.
struct WaveCtx {
  int tid, lane, lr, lh, mw, nw;
};
static __device__ __forceinline__ WaveCtx make_ctx() {
  WaveCtx c;
  c.tid  = threadIdx.x;
  c.lane = c.tid & 31;
  c.lr   = c.lane & 15;
  c.lh   = c.lane >> 4;
  const int wid = c.tid >> 5;
  c.mw = (wid >> 1) * 32;
  c.nw = (wid & 1) * 64;
  return c;
}

static __device__ __forceinline__ void mma_stage(
    const float* __restrict__ sA, const float* __restrict__ sB,
    const WaveCtx& c, v8f acc[2][4]) {
#pragma unroll
  for (int kk = 0; kk < KT; kk += 4) {
    const int kof = kk + c.lh * 2;
    v2f a0 = *(const v2f*)&sA[(c.mw + c.lr) * LDA + kof];
    v2f a1 = *(const v2f*)&sA[(c.mw + 16 + c.lr) * LDA + kof];
#pragma unroll
    for (int j = 0; j < 4; ++j) {
      const int col = c.nw + j * 16 + c.lr;
      v2f bf;
      bf.x = sB[kof * LDB + col];
      bf.y = sB[(kof + 1) * LDB + col];
      acc[0][j] = wmma_f32_16x16x4(a0, bf, acc[0][j]);
      acc[1][j] = wmma_f32_16x16x4(a1, bf, acc[1][j]);
    }
  }
}

// Issue one stage of async copies: 4 x b64 per thread for A, 4 for B.
static __device__ __forceinline__ void issue_a(const float* __restrict__ gA,
                                               int ldg, float* sA, int tid) {
#pragma unroll
  for (int p = 0; p < 4; ++p) {
    const int f2 = p * 256 + tid;
    const int r = f2 >> 3, c2 = f2 & 7;                    // 128 rows x 8 float2
    async_copy_b64(&sA[r * LDA + c2 * 2], gA + (size_t)r * ldg + c2 * 2);
  }
}
static __device__ __forceinline__ void issue_b(const float* __restrict__ gB,
                                               int ldg, float* sB, int tid) {
#pragma unroll
  for (int p = 0; p < 4; ++p) {
    const int f2 = p * 256 + tid;
    const int r = f2 >> 6, c2 = f2 & 63;                   // KT rows x 64 float2
    async_copy_b64(&sB[r * LDB + c2 * 2], gB + (size_t)r * ldg + c2 * 2);
  }
}

// -------------------------------------------- projection: hs = D^-1/2 (xW^T+b)
__global__ __launch_bounds__(256) void gcn_proj_kernel(
    const float* __restrict__ x, const float* __restrict__ Wt,
    const float* __restrict__ bias, const float* __restrict__ dinv,
    float* __restrict__ hs) {
  __shared__ float sA[2][ASZ];
  __shared__ float sB[2][BSZ];
  const WaveCtx c = make_ctx();
  const int m0 = blockIdx.x * 128;                         // row block over B*N
  const int n0 = blockIdx.y * 128;                         // output-feature block

  const v8f vz = {0.f, 0.f, 0.f, 0.f, 0.f, 0.f, 0.f, 0.f};
  v8f acc[2][4];
#pragma unroll
  for (int g = 0; g < 2; ++g)
#pragma unroll
    for (int j = 0; j < 4; ++j) acc[g][j] = vz;

  const float* __restrict__ gA0 = x + (size_t)m0 * DD;
  const int S = DD / KT;
  issue_a(gA0, DD, sA[0], c.tid);
  issue_b(Wt + n0, DD, sB[0], c.tid);
  for (int s = 0; s < S; ++s) {
    const int cur = s & 1;
    if (s + 1 < S) {
      issue_a(gA0 + (s + 1) * KT, DD, sA[cur ^ 1], c.tid);
      issue_b(Wt + (size_t)(s + 1) * KT * DD + n0, DD, sB[cur ^ 1], c.tid);
      asm volatile("s_wait_asynccnt 0x8" ::: "memory");    // stage s complete
    } else {
      asm volatile("s_wait_asynccnt 0x0" ::: "memory");
    }
    __syncthreads();
    mma_stage(sA[cur], sB[cur], c, acc);
    __syncthreads();   // all waves done reading buf[cur] before it is refilled
  }

#pragma unroll
  for (int g = 0; g < 2; ++g) {
    const int rbase = m0 + c.mw + g * 16 + c.lh * 8;
#pragma unroll
    for (int j = 0; j < 4; ++j) {
      const int col = n0 + c.nw + j * 16 + c.lr;
      const float bc = bias[col];
#pragma unroll
      for (int i = 0; i < 8; ++i) {
        const int row = rbase + i;
        hs[(size_t)row * DD + col] = dinv[row] * (acc[g][j][i] + bc);
      }
    }
  }
}

// ----------------------------------------- aggregation: out = D^-1/2 A @ hs --
__global__ __launch_bounds__(256) void gcn_agg_kernel(
    const float* __restrict__ adj, const float* __restrict__ hs,
    const float* __restrict__ dinv, float* __restrict__ out) {
  __shared__ float sA[2][ASZ];
  __shared__ float sB[2][BSZ];
  const WaveCtx c = make_ctx();
  const int b  = blockIdx.z;
  const int m0 = blockIdx.x * 128;
  const int n0 = blockIdx.y * 128;
  const float* __restrict__ adjb = adj + (size_t)b * NN * NN;
  const float* __restrict__ hsb  = hs + (size_t)b * NN * DD;
  const float* __restrict__ dvb  = dinv + (size_t)b * NN;
  float* __restrict__ outb       = out + (size_t)b * NN * DD;

  const v8f vz = {0.f, 0.f, 0.f, 0.f, 0.f, 0.f, 0.f, 0.f};
  v8f acc[2][4];
#pragma unroll
  for (int g = 0; g < 2; ++g)
#pragma unroll
    for (int j = 0; j < 4; ++j) acc[g][j] = vz;

  const float* __restrict__ gA0 = adjb + (size_t)m0 * NN;
  const int S = NN / KT;
  issue_a(gA0, NN, sA[0], c.tid);
  issue_b(hsb + n0, DD, sB[0], c.tid);
  for (int s = 0; s < S; ++s) {
    const int cur = s & 1;
    if (s + 1 < S) {
      issue_a(gA0 + (s + 1) * KT, NN, sA[cur ^ 1], c.tid);
      issue_b(hsb + (size_t)(s + 1) * KT * DD + n0, DD, sB[cur ^ 1], c.tid);
      asm volatile("s_wait_asynccnt 0x8" ::: "memory");    // stage s complete
    } else {
      asm volatile("s_wait_asynccnt 0x0" ::: "memory");
    }
    __syncthreads();
    mma_stage(sA[cur], sB[cur], c, acc);
    __syncthreads();   // all waves done reading buf[cur] before it is refilled
  }

#pragma unroll
  for (int g = 0; g < 2; ++g) {
    const int rbase = m0 + c.mw + g * 16 + c.lh * 8;
#pragma unroll
    for (int j = 0; j < 4; ++j) {
      const int col = n0 + c.nw + j * 16 + c.lr;
#pragma unroll
      for (int i = 0; i < 8; ++i) {
        const int row = rbase + i;
        outb[(size_t)row * DD + col] = dvb[row] * acc[g][j][i];
      }
    }
  }
}

// ------------------------------------------------------------------ launch ---
extern "C" void kernel_launch(void* const* d_in, const int* in_sizes, int n_in,
                              void* d_out, int out_size, void* d_ws, size_t ws_size,
                              hipStream_t stream) {
  const float* x    = (const float*)d_in[0];   // [B,N,256]
  const float* adj  = (const float*)d_in[1];   // [B,N,N]
  const float* W    = (const float*)d_in[2];   // [256,256]
  const float* bias = (const float*)d_in[3];   // [256]
  float* out = (float*)d_out;                  // [B,N,256]

  float* dinv = (float*)d_ws;                  // B*N floats
  float* Wt   = dinv + (size_t)BATCH * NN;     // D*D floats
  float* hs   = Wt + (size_t)DD * DD;          // B*N*D floats

  gcn_degree_kernel<<<dim3(BATCH * NN / 8), dim3(32, 8), 0, stream>>>(adj, dinv);
  gcn_transpose_kernel<<<dim3(DD / 32, DD / 32), dim3(32, 8), 0, stream>>>(W, Wt);
  gcn_proj_kernel<<<dim3(BATCH * NN / 128, DD / 128), dim3(256), 0, stream>>>(
      x, Wt, bias, dinv, hs);
  gcn_agg_kernel<<<dim3(NN / 128, DD / 128, BATCH), dim3(256), 0, stream>>>(
      adj, hs, dinv, out);
}